// Net_nodes_23398981828940
// MI455X (gfx1250) — compile-verified
//
#include <hip/hip_runtime.h>

typedef float v2f __attribute__((ext_vector_type(2)));
typedef float v8f __attribute__((ext_vector_type(8)));

#define NN   25          // nodes
#define NP   32          // padded nodes
#define CH   64          // channels
#define NE   64          // edges per graph
#define NSLICE 19200     // B*T = 64*300

// ws float layout
#define AOP_OFF  0                    // [3][2][7][32] v2f   = 2688 floats
#define BSUM_OFF 2688                 // [64] float
#define WOP_OFF  2752                 // [3][16][2][64] v2f  = 12288 floats

// D = A(16x4,f32) * B(4x16,f32) + C(16x16,f32)
static __device__ __forceinline__ v8f wmma4(v2f a, v2f b, v8f c) {
    return __builtin_amdgcn_wmma_f32_16x16x4_f32(false, a, false, b, (short)0, c,
                                                 false, false);
}

static __device__ __forceinline__ v8f splat8(float f) {
    return (v8f){f, f, f, f, f, f, f, f};
}

// ---------------------------------------------------------------------------
// Kernel 1: build normalized adjacency, pack A and W into WMMA operand layout
// ---------------------------------------------------------------------------
__global__ void __launch_bounds__(128)
build_ops_kernel(const int* __restrict__ ei,     // [3][2][64]
                 const float* __restrict__ W1,
                 const float* __restrict__ W2,
                 const float* __restrict__ W3,
                 const float* __restrict__ b1,
                 const float* __restrict__ b2,
                 const float* __restrict__ b3,
                 float* __restrict__ ws)
{
    __shared__ float Ash[NP][NP];
    __shared__ float deg[NP];
    const int t = threadIdx.x;
    const float* Wk[3] = {W1, W2, W3};

    for (int k = 0; k < 3; ++k) {
        const int* row = ei + k * 128;       // sources
        const int* col = ei + k * 128 + 64;  // destinations
        if (t < NP) deg[t] = 0.0f;
        for (int i = t; i < NP * NP; i += 128) (&Ash[0][0])[i] = 0.0f;
        __syncthreads();
        if (t < NE) atomicAdd(&deg[col[t]], 1.0f);
        if (t < NN) atomicAdd(&deg[t], 1.0f);          // self loop
        __syncthreads();
        if (t < NP) deg[t] = (deg[t] > 0.0f) ? rsqrtf(deg[t]) : 0.0f;  // -> dinv
        __syncthreads();
        if (t < NE) {
            int r = row[t], c = col[t];
            atomicAdd(&Ash[c][r], deg[r] * deg[c]);    // A[dst][src] += norm
        }
        if (t < NN) atomicAdd(&Ash[t][t], deg[t] * deg[t]);
        __syncthreads();
        // pack A operands for this k: pair p = ((k*2+mt)*7+ks)*32 + lane
        //   value = { A[16mt+ln][4ks+2h], A[16mt+ln][4ks+2h+1] }
        for (int i = t; i < 2 * 7 * 32; i += 128) {
            int mt = i / (7 * 32);
            int r2 = i % (7 * 32);
            int ks = r2 >> 5;
            int lane = r2 & 31;
            int ln = lane & 15, half = lane >> 4;
            int n = mt * 16 + ln;
            int m = ks * 4 + half * 2;
            int p = ((k * 2 + mt) * 7 + ks) * 32 + lane;
            ws[AOP_OFF + 2 * p]     = Ash[n][m];
            ws[AOP_OFF + 2 * p + 1] = Ash[n][m + 1];
        }
        __syncthreads();
    }

    // pack W operands: pair p = ((k*16+ks)*2+half)*64 + d
    //   value = { W_k[4ks+2h][d], W_k[4ks+2h+1][d] }
    for (int i = t; i < 3 * 16 * 2 * 64; i += 128) {
        int k  = i >> 11;
        int r  = i & 2047;
        int ks = r >> 7;
        int r2 = r & 127;
        int half = r2 >> 6;
        int d  = r2 & 63;
        int c0 = ks * 4 + half * 2;
        ws[WOP_OFF + 2 * i]     = Wk[k][c0 * CH + d];
        ws[WOP_OFF + 2 * i + 1] = Wk[k][(c0 + 1) * CH + d];
    }
    if (t < CH) ws[BSUM_OFF + t] = b1[t] + b2[t] + b3[t];
}

// ---------------------------------------------------------------------------
// Kernel 2: fused  out[s] = sum_k A_k * (X[s] * W_k) + bsum,  one wave / slice
// ---------------------------------------------------------------------------
__global__ void __launch_bounds__(128)
gcn_main_kernel(const float* __restrict__ x,      // [19200][25][64]
                const v2f* __restrict__ Aop,      // packed, see kernel 1
                const v2f* __restrict__ Wop,      // packed, see kernel 1
                const float* __restrict__ bsum,   // [64]
                float* __restrict__ out)          // [19200][25][64]
{
    __shared__ float zbuf[4][NP][68];   // per-wave Z staging, bank-pad stride 68

    const int lane = threadIdx.x & 31;
    const int wv   = threadIdx.x >> 5;
    const int half = lane >> 4;     // 0: lanes 0-15, 1: lanes 16-31
    const int ln   = lane & 15;

    float bb[4];
#pragma unroll
    for (int nt = 0; nt < 4; ++nt) bb[nt] = bsum[nt * 16 + ln];

    const int nWaves = gridDim.x * 4;
    for (int s = blockIdx.x * 4 + wv; s < NSLICE; s += nWaves) {
        const float* xs = x + (size_t)s * (NN * CH);

        // X in B-operand layout: b.v0 = X[4ks+2h, c], b.v1 = X[4ks+2h+1, c]
        v2f xb[7][4];
#pragma unroll
        for (int ks = 0; ks < 6; ++ks) {              // rows 0..23: all valid
            const int m0 = ks * 4 + half * 2;
#pragma unroll
            for (int nt = 0; nt < 4; ++nt) {
                const int c = nt * 16 + ln;
                float e0 = __builtin_nontemporal_load(xs + m0 * CH + c);
                float e1 = __builtin_nontemporal_load(xs + (m0 + 1) * CH + c);
                xb[ks][nt] = (v2f){e0, e1};
            }
        }
        // ks = 6: rows {24,25 | 26,27}; only row 24 (low half, v0) is real.
        // Branchless: load row 24 unconditionally, mask with cndmask.
#pragma unroll
        for (int nt = 0; nt < 4; ++nt) {
            const int c = nt * 16 + ln;
            float e0 = __builtin_nontemporal_load(xs + 24 * CH + c);
            xb[6][nt] = (v2f){half ? 0.0f : e0, 0.0f};
        }

        // bias pre-folded: every element of tile (mt,nt) gets bias(col)=bb[nt]
        v8f acc[2][4];
#pragma unroll
        for (int mt = 0; mt < 2; ++mt)
#pragma unroll
            for (int nt = 0; nt < 4; ++nt) acc[mt][nt] = splat8(bb[nt]);

#pragma unroll
        for (int k = 0; k < 3; ++k) {
            // ---- phase 1: Z_k = A_k @ X   (M=32, K=28, N=64) ----
            v8f z[2][4];
#pragma unroll
            for (int mt = 0; mt < 2; ++mt)
#pragma unroll
                for (int nt = 0; nt < 4; ++nt) {
                    v8f zz = (v8f)(0.0f);
#pragma unroll
                    for (int ks = 0; ks < 7; ++ks) {
                        v2f a = Aop[((k * 2 + mt) * 7 + ks) * 32 + lane];
                        zz = wmma4(a, xb[ks][nt], zz);
                    }
                    z[mt][nt] = zz;
                }
            // stage Z: C-layout (VGPR v -> M = v + 8*half, lane -> N) to row-major
#pragma unroll
            for (int mt = 0; mt < 2; ++mt)
#pragma unroll
                for (int nt = 0; nt < 4; ++nt)
#pragma unroll
                    for (int v = 0; v < 8; ++v)
                        zbuf[wv][mt * 16 + half * 8 + v][nt * 16 + ln] = z[mt][nt][v];

            // ---- phase 2: acc += Z_k @ W_k   (M=32, K=64, N=64) ----
#pragma unroll
            for (int ks = 0; ks < 16; ++ks) {
                const int c0 = ks * 4 + half * 2;
                v2f za[2];
#pragma unroll
                for (int mt = 0; mt < 2; ++mt)
                    za[mt] = *(const v2f*)&zbuf[wv][mt * 16 + ln][c0];
#pragma unroll
                for (int nt = 0; nt < 4; ++nt) {
                    v2f wb = Wop[((k * 16 + ks) * 2 + half) * 64 + nt * 16 + ln];
#pragma unroll
                    for (int mt = 0; mt < 2; ++mt)
                        acc[mt][nt] = wmma4(za[mt], wb, acc[mt][nt]);
                }
            }
        }

        // write back (skip padded rows); bias already folded in
        float* os = out + (size_t)s * (NN * CH);
#pragma unroll
        for (int mt = 0; mt < 2; ++mt)
#pragma unroll
            for (int nt = 0; nt < 4; ++nt)
#pragma unroll
                for (int v = 0; v < 8; ++v) {
                    const int n = mt * 16 + half * 8 + v;
                    if (n < NN)
                        __builtin_nontemporal_store(acc[mt][nt][v],
                                                    os + n * CH + nt * 16 + ln);
                }
    }
}

// ---------------------------------------------------------------------------
extern "C" void kernel_launch(void* const* d_in, const int* in_sizes, int n_in,
                              void* d_out, int out_size, void* d_ws, size_t ws_size,
                              hipStream_t stream)
{
    const float* x  = (const float*)d_in[0];
    const int*   ei = (const int*)d_in[1];
    const float* W1 = (const float*)d_in[2];
    const float* W2 = (const float*)d_in[3];
    const float* W3 = (const float*)d_in[4];
    const float* b1 = (const float*)d_in[5];
    const float* b2 = (const float*)d_in[6];
    const float* b3 = (const float*)d_in[7];
    float* outp = (float*)d_out;
    float* ws   = (float*)d_ws;

    build_ops_kernel<<<1, 128, 0, stream>>>(ei, W1, W2, W3, b1, b2, b3, ws);
    gcn_main_kernel<<<1200, 128, 0, stream>>>(x,
                                              (const v2f*)(ws + AOP_OFF),
                                              (const v2f*)(ws + WOP_OFF),
                                              ws + BSUM_OFF,
                                              outp);
}